// Floyd_37091337568318
// MI455X (gfx1250) — compile-verified
//
#include <hip/hip_runtime.h>
#include <math.h>

// ---------------------------------------------------------------------------
// Floyd-style cycle-consistent multi-matching refinement.
// m=16, n=32.  K: 256 blocks of 1024x1024 f32 (1 GiB).  X: 256 perm matrices.
//
// Roofline: affinity vx^T K vx with one-hot vx (32 ones) == sum of a 32x32
// gathered submatrix of K.  Sparse gather: <=1024 cachelines (128B) per pair
// per step -> <=32MB/step, 33 steps ~ 1GB ~ 45us @ 23.3TB/s.  Dense WMMA on K
// would stream 33 GiB (~1.4ms).  So: gather in f32, use WMMA f32 16x16x4 for
// the in-wave reduction (ones-matrix trick), keep all scoring in f32.
// ---------------------------------------------------------------------------

typedef __attribute__((ext_vector_type(2))) float v2f;
typedef __attribute__((ext_vector_type(8))) float v8f;

#define MM 16
#define NP 32          // n = 32
#define PAIRS 256      // m*m

// ---- perm extraction: perm[b][a] = argmax_c X[b][a][c] ---------------------
__global__ __launch_bounds__(32)
void extract_kernel(const float* __restrict__ X, int* __restrict__ perm) {
  int b = blockIdx.x;
  int a = threadIdx.x;
  const float* row = X + ((size_t)b * NP + a) * NP;
  int c = 0;
  #pragma unroll
  for (int k = 0; k < NP; ++k)
    if (row[k] > 0.5f) c = k;
  perm[b * NP + a] = c;
}

// ---- pairwise consistency: pc[i,j] = 1 - sum_k ||X_ik X_kj - X_ij||_F /1024
__global__ __launch_bounds__(32)
void pc_kernel(const int* __restrict__ perm, float* __restrict__ pc) {
  int b = blockIdx.x;
  int i = b >> 4, j = b & 15;
  int a = threadIdx.x;                    // one wave32
  int target = perm[b * NP + a];
  float acc = 0.0f;
  for (int k = 0; k < MM; ++k) {
    int p1 = perm[(i * MM + k) * NP + a];
    int p2 = perm[(k * MM + j) * NP + p1];
    unsigned long long mask = __ballot(p2 == target);
    int agree = __popcll(mask);
    // ||P - Q||_F^2 = 2*(n - agree) for permutation matrices
    acc += sqrtf(2.0f * (float)(NP - agree) + 1e-12f);
  }
  if (a == 0) pc[b] = 1.0f - acc * (1.0f / 1024.0f);   // 2*n*m = 1024
}

// ---- wave-level gather + WMMA reduction ------------------------------------
// Wave w of a 256-thread block owns rows 4w..4w+3 of the gathered 32x32
// submatrix G[t,u] = Kb[q_t*1024 + q_u].  128 values are packed as two 16x4
// f32 A-tiles; B = ones(4x16); D accumulates row sums (replicated across N).
// D column 0 lives in lanes 0 (M=0..7) and 16 (M=8..15).
__device__ __forceinline__ float wave_gather_wmma_sum(
    const float* __restrict__ Kb, const int* __restrict__ q, int w, int l) {
  int qc = q[l];
  float v0 = Kb[(size_t)q[4 * w + 0] * 1024 + qc];
  float v1 = Kb[(size_t)q[4 * w + 1] * 1024 + qc];
  float v2 = Kb[(size_t)q[4 * w + 2] * 1024 + qc];
  float v3 = Kb[(size_t)q[4 * w + 3] * 1024 + qc];
  v2f a1 = {v0, v1};
  v2f a2 = {v2, v3};
  v2f bone = {1.0f, 1.0f};
  v8f c = {};
  c = __builtin_amdgcn_wmma_f32_16x16x4_f32(false, a1, false, bone,
                                            (short)0, c, false, false);
  c = __builtin_amdgcn_wmma_f32_16x16x4_f32(false, a2, false, bone,
                                            (short)0, c, false, false);
  float s = c[0] + c[1] + c[2] + c[3] + c[4] + c[5] + c[6] + c[7];
  float shi = __shfl(s, 16, 32);          // rows 8..15 partial from lane 16
  return s + shi;                         // valid on lanes 0..15
}

// ---- initial score: sc[b] = 0.3*pc[b] + 0.7*aff[b] -------------------------
__global__ __launch_bounds__(256)
void aff_init_kernel(const float* __restrict__ K, const int* __restrict__ perm,
                     const float* __restrict__ pc, float* __restrict__ sc) {
  int b = blockIdx.x;
  __shared__ int q[NP];
  __shared__ float wsum[8];
  int tid = threadIdx.x;
  if (tid < NP) q[tid] = perm[b * NP + tid] * NP + tid;   // q_t = sigma(t)*32+t
  __syncthreads();
  const float* Kb = K + (size_t)b * (1024u * 1024u);
  int w = tid >> 5, l = tid & 31;
  float ws = wave_gather_wmma_sum(Kb, q, w, l);
  if (l == 0) wsum[w] = ws;
  __syncthreads();
  if (tid == 0) {
    float S = 0.0f;
    #pragma unroll
    for (int i = 0; i < 8; ++i) S += wsum[i];
    float aff = S * (1.0f / 32.0f);
    sc[b] = 0.3f * pc[b] + 0.7f * aff;
  }
}

// ---- snapshot row v and column v of the current perms ----------------------
__global__ __launch_bounds__(512)
void snapshot_kernel(const int* __restrict__ perm, int* __restrict__ snap1,
                     int* __restrict__ snap2, int vc) {
  int t = threadIdx.x;            // 512 = 16*32
  int i = t >> 5, a = t & 31;
  snap1[i * NP + a] = perm[(i * MM + vc) * NP + a];   // X1 = Xc[:, v]
  snap2[i * NP + a] = perm[(vc * MM + i) * NP + a];   // X2 = Xc[v, :]
}

// ---- one Floyd step: candidate sigma_u = X2_j o X1_i, accept if better -----
__global__ __launch_bounds__(256)
void aff_update_kernel(const float* __restrict__ K, int* __restrict__ perm,
                       const float* __restrict__ pc, float* __restrict__ sc,
                       const int* __restrict__ snap1,
                       const int* __restrict__ snap2, int vc) {
  int b = blockIdx.x;
  int i = b >> 4, j = b & 15;
  __shared__ int q[NP];
  __shared__ int sig[NP];
  __shared__ float wsum[8];
  __shared__ int accept;
  int tid = threadIdx.x;
  if (tid < NP) {
    int p1 = snap1[i * NP + tid];
    int su = snap2[j * NP + p1];            // sigma_u(a) = X2_j[X1_i[a]]
    sig[tid] = su;
    q[tid] = su * NP + tid;
  }
  __syncthreads();
  const float* Kb = K + (size_t)b * (1024u * 1024u);
  int w = tid >> 5, l = tid & 31;
  float ws = wave_gather_wmma_sum(Kb, q, w, l);
  if (l == 0) wsum[w] = ws;
  __syncthreads();
  if (tid == 0) {
    float S = 0.0f;
    #pragma unroll
    for (int k = 0; k < 8; ++k) S += wsum[k];
    float aff = S * (1.0f / 32.0f);
    float pcu = sqrtf(pc[i * MM + vc] * pc[vc * MM + j]);
    float scu = 0.3f * pcu + 0.7f * aff;
    accept = (scu > sc[b]) ? 1 : 0;
    if (accept) sc[b] = scu;
  }
  __syncthreads();
  if (accept && tid < NP) perm[b * NP + tid] = sig[tid];
}

// ---- write back one-hot matrices -------------------------------------------
__global__ __launch_bounds__(1024)
void writeout_kernel(const int* __restrict__ perm, float* __restrict__ out) {
  int b = blockIdx.x;
  int t = threadIdx.x;
  int a = t >> 5, c = t & 31;
  out[((size_t)b * NP + a) * NP + c] = (perm[b * NP + a] == c) ? 1.0f : 0.0f;
}

extern "C" void kernel_launch(void* const* d_in, const int* in_sizes, int n_in,
                              void* d_out, int out_size, void* d_ws,
                              size_t ws_size, hipStream_t stream) {
  const float* K = (const float*)d_in[0];   // (16,16,1024,1024) f32
  const float* X = (const float*)d_in[1];   // (16,16,32,32) f32
  float* out = (float*)d_out;               // (16,16,32,32) f32

  int*   perm  = (int*)d_ws;                // 256*32 ints
  int*   snap1 = perm + PAIRS * NP;         // 16*32
  int*   snap2 = snap1 + MM * NP;           // 16*32
  float* pc    = (float*)(snap2 + MM * NP); // 256
  float* sc    = pc + PAIRS;                // 256

  extract_kernel<<<PAIRS, 32, 0, stream>>>(X, perm);
  pc_kernel<<<PAIRS, 32, 0, stream>>>(perm, pc);
  aff_init_kernel<<<PAIRS, 256, 0, stream>>>(K, perm, pc, sc);

  for (int v = 0; v < NP; ++v) {            // reference scans arange(n)=0..31
    int vc = (v < MM) ? v : (MM - 1);       // JAX clamps OOB index to 15
    snapshot_kernel<<<1, 512, 0, stream>>>(perm, snap1, snap2, vc);
    aff_update_kernel<<<PAIRS, 256, 0, stream>>>(K, perm, pc, sc, snap1,
                                                 snap2, vc);
  }
  writeout_kernel<<<PAIRS, 1024, 0, stream>>>(perm, out);
}